// DictionaryExpertLISTA_81647328297255
// MI455X (gfx1250) — compile-verified
//
#include <hip/hip_runtime.h>
#include <hip/hip_bf16.h>

// ---------------------------------------------------------------------------
// LISTA sparse coder for MI455X (gfx1250, wave32).
//   per layer: P = x @ W      (dense f32 WMMA GEMM, V_WMMA_F32_16X16X4_F32)
//              u = P + z @ S  (sparse gather: z has <=64 nnz/row)
//              z = top64(|u|) (bisection threshold + deterministic compaction)
//   recon = z @ dict^T        (sparse gather against pre-transposed dict)
// ---------------------------------------------------------------------------

typedef float v2f __attribute__((ext_vector_type(2)));
typedef float v8f __attribute__((ext_vector_type(8)));

#define BATCH   8192
#define IN_DIM  1024
#define CODE    4096
#define NLAYERS 5
#define KSP     64

// ---------------------------------------------------------------------------
__global__ void zero_counts_kernel(int* __restrict__ counts, int n) {
  int i = blockIdx.x * blockDim.x + threadIdx.x;
  if (i < n) counts[i] = 0;
}

// dict [1024][4096] -> Dt [4096][1024]  (so recon gathers coalesced rows)
__global__ __launch_bounds__(256)
void transpose_dict_kernel(const float* __restrict__ D, float* __restrict__ Dt) {
  __shared__ float tile[32][33];
  int tx = threadIdx.x & 31;
  int ty = threadIdx.x >> 5;           // 0..7
  int c0 = blockIdx.x * 32;            // along CODE
  int r0 = blockIdx.y * 32;            // along IN_DIM
#pragma unroll
  for (int k = 0; k < 4; ++k)
    tile[ty + 8 * k][tx] = D[(size_t)(r0 + ty + 8 * k) * CODE + (c0 + tx)];
  __syncthreads();
#pragma unroll
  for (int k = 0; k < 4; ++k)
    Dt[(size_t)(c0 + ty + 8 * k) * IN_DIM + (r0 + tx)] = tile[tx][ty + 8 * k];
}

// ---------------------------------------------------------------------------
// P[8192x4096] = x[8192x1024] @ W[1024x4096], f32 WMMA 16x16x4.
//
// Register-blocked: each wave computes a 64x32 C macro-tile as 4(M) x 2(N)
// 16x16 WMMA accumulators.  Per K-step of 4: 4 b64 A-loads + 4 b32 B-loads
// feed 8 WMMAs  ->  ~1 vmem instruction per WMMA (vs 3 unblocked).
// Block = 8 waves arranged 2(M) x 4(N)  ->  128x128 block tile; the 4 waves in
// an N-row share identical A rows (WGP$ reuse).
//
// Fragment layouts (CDNA5 ISA 7.12.2, 32-bit):
//   A 16x4 : lane&15 = M; VGPR0 holds K=(lane>>4)*2, VGPR1 holds K+1
//   B 4x16 : lane&15 = N; VGPR0 holds K=(lane>>4)*2 row, VGPR1 holds K+1 row
//   C 16x16: VGPR r -> row r (lanes 0-15) / row r+8 (lanes 16-31), N=lane&15
__global__ __launch_bounds__(256)
void gemm_xw_wmma(const float* __restrict__ A, const float* __restrict__ B,
                  float* __restrict__ C) {
  const int N = CODE, Klen = IN_DIM;
  int lane  = threadIdx.x & 31;
  int wave  = threadIdx.x >> 5;
  int waveN = wave & 3;                         // 0..3
  int waveM = wave >> 2;                        // 0..1
  int laneM = lane & 15;
  int hi    = lane >> 4;                        // 0 or 1

  int R  = blockIdx.y * 128 + waveM * 64;       // first row of wave macro-tile
  int Cb = blockIdx.x * 128 + waveN * 32;       // first col of wave macro-tile

  const float* ap = A + (size_t)(R + laneM) * Klen + (hi << 1);
  const float* bp = B + (size_t)(hi << 1) * N + (Cb + laneM);

  v8f acc[4][2];
#pragma unroll
  for (int m = 0; m < 4; ++m)
#pragma unroll
    for (int n = 0; n < 2; ++n)
      acc[m][n] = (v8f){0.f, 0.f, 0.f, 0.f, 0.f, 0.f, 0.f, 0.f};

#pragma unroll 2
  for (int k = 0; k < Klen; k += 4) {
    v2f a[4];
#pragma unroll
    for (int m = 0; m < 4; ++m)                 // rows R+16m (64KB imm offsets)
      a[m] = *(const v2f*)(ap + (size_t)m * 16 * Klen);
    v2f b[2];
#pragma unroll
    for (int n = 0; n < 2; ++n) {               // cols Cb+16n (64B imm offsets)
      b[n].x = bp[n * 16];                      // B row K
      b[n].y = bp[n * 16 + N];                  // B row K+1
    }
#pragma unroll
    for (int m = 0; m < 4; ++m)
#pragma unroll
      for (int n = 0; n < 2; ++n)
        acc[m][n] = __builtin_amdgcn_wmma_f32_16x16x4_f32(
            false, a[m], false, b[n], (short)0, acc[m][n], false, false);
    ap += 4;
    bp += (size_t)4 * N;
  }

#pragma unroll
  for (int m = 0; m < 4; ++m)
#pragma unroll
    for (int n = 0; n < 2; ++n) {
      float* cp = C + (size_t)(R + m * 16 + hi * 8) * N + (Cb + n * 16 + laneM);
#pragma unroll
      for (int r = 0; r < 8; ++r)
        cp[(size_t)r * N] = acc[m][n][r];
    }
}

// ---------------------------------------------------------------------------
// One block per batch row:
//   u = P[b,:] + sum_n vals[n]*S[idxs[n],:]
//   thr = top-64 |u| threshold via 40-step bisection
//   deterministic (column-ordered) compaction into idxs/vals; dense z on last layer
__global__ __launch_bounds__(256)
void lista_step_kernel(const float* __restrict__ P, const float* __restrict__ S,
                       int* __restrict__ idxs, float* __restrict__ vals,
                       int* __restrict__ counts, float* __restrict__ zOut,
                       int writeZ) {
  int b = blockIdx.x;
  int t = threadIdx.x;                          // 0..255, owns cols [16t,16t+16)
  __shared__ int   s_idx[KSP];
  __shared__ float s_val[KSP];
  __shared__ float s_fred[256];
  __shared__ int   s_ired[256];

  int cnt = counts[b];
  if (t < cnt) { s_idx[t] = idxs[b * KSP + t]; s_val[t] = vals[b * KSP + t]; }
  __syncthreads();

  float u[16];
  const float* prow = P + (size_t)b * CODE + t * 16;
#pragma unroll
  for (int v = 0; v < 4; ++v) {
    float4 q = ((const float4*)prow)[v];
    u[4 * v + 0] = q.x; u[4 * v + 1] = q.y;
    u[4 * v + 2] = q.z; u[4 * v + 3] = q.w;
  }
  for (int n = 0; n < cnt; ++n) {
    float sv = s_val[n];
    const float* srow = S + (size_t)s_idx[n] * CODE + t * 16;
#pragma unroll
    for (int v = 0; v < 4; ++v) {
      float4 q = ((const float4*)srow)[v];
      u[4 * v + 0] = fmaf(sv, q.x, u[4 * v + 0]);
      u[4 * v + 1] = fmaf(sv, q.y, u[4 * v + 1]);
      u[4 * v + 2] = fmaf(sv, q.z, u[4 * v + 2]);
      u[4 * v + 3] = fmaf(sv, q.w, u[4 * v + 3]);
    }
  }

  // --- max |u| across the row ---
  float m = 0.f;
#pragma unroll
  for (int i = 0; i < 16; ++i) m = fmaxf(m, fabsf(u[i]));
  s_fred[t] = m;
  __syncthreads();
  for (int s = 128; s > 0; s >>= 1) {
    if (t < s) s_fred[t] = fmaxf(s_fred[t], s_fred[t + s]);
    __syncthreads();
  }
  float lo = 0.f, hi = s_fred[0];
  __syncthreads();

  // --- bisection: keep count(|u|>lo) >= 64 > count(|u|>hi) ---
  for (int it = 0; it < 40; ++it) {
    float mid = 0.5f * (lo + hi);
    int c = 0;
#pragma unroll
    for (int i = 0; i < 16; ++i) c += (fabsf(u[i]) > mid) ? 1 : 0;
    s_ired[t] = c;
    __syncthreads();
    for (int s = 128; s > 0; s >>= 1) {
      if (t < s) s_ired[t] += s_ired[t + s];
      __syncthreads();
    }
    int total = s_ired[0];
    __syncthreads();
    if (total >= KSP) lo = mid; else hi = mid;
  }
  float thr = lo;

  // --- deterministic compaction: exclusive scan of per-thread keep counts ---
  int kcnt = 0;
#pragma unroll
  for (int i = 0; i < 16; ++i) kcnt += (fabsf(u[i]) > thr) ? 1 : 0;
  s_ired[t] = kcnt;
  __syncthreads();
  for (int s = 1; s < 256; s <<= 1) {          // Hillis-Steele inclusive scan
    int v = (t >= s) ? s_ired[t - s] : 0;
    __syncthreads();
    s_ired[t] += v;
    __syncthreads();
  }
  int pos   = s_ired[t] - kcnt;                // exclusive prefix (column order)
  int total = s_ired[255];

#pragma unroll
  for (int i = 0; i < 16; ++i) {
    bool keep = fabsf(u[i]) > thr;
    if (keep) {
      if (pos < KSP) { idxs[b * KSP + pos] = t * 16 + i; vals[b * KSP + pos] = u[i]; }
      ++pos;
    }
    if (!keep) u[i] = 0.f;                     // masked value for dense z
  }
  if (writeZ) {
    float* zrow = zOut + (size_t)b * CODE + t * 16;
#pragma unroll
    for (int v = 0; v < 4; ++v) {
      float4 q;
      q.x = u[4 * v + 0]; q.y = u[4 * v + 1];
      q.z = u[4 * v + 2]; q.w = u[4 * v + 3];
      ((float4*)zrow)[v] = q;
    }
  }
  if (t == 0) counts[b] = (total < KSP) ? total : KSP;
}

// ---------------------------------------------------------------------------
// recon[b,:] = sum_n vals[n] * Dt[idxs[n],:]   (Dt = dict^T, rows coalesced)
__global__ __launch_bounds__(256)
void recon_kernel(const float* __restrict__ Dt, const int* __restrict__ idxs,
                  const float* __restrict__ vals, const int* __restrict__ counts,
                  float* __restrict__ recon) {
  int b = blockIdx.x;
  int t = threadIdx.x;                          // owns cols [4t, 4t+4)
  __shared__ int   s_idx[KSP];
  __shared__ float s_val[KSP];
  int cnt = counts[b];
  if (t < cnt) { s_idx[t] = idxs[b * KSP + t]; s_val[t] = vals[b * KSP + t]; }
  __syncthreads();

  float r0 = 0.f, r1 = 0.f, r2 = 0.f, r3 = 0.f;
  for (int n = 0; n < cnt; ++n) {
    float sv = s_val[n];
    const float* drow = Dt + (size_t)s_idx[n] * IN_DIM + t * 4;
    float4 q = *(const float4*)drow;
    r0 = fmaf(sv, q.x, r0); r1 = fmaf(sv, q.y, r1);
    r2 = fmaf(sv, q.z, r2); r3 = fmaf(sv, q.w, r3);
  }
  float4 out; out.x = r0; out.y = r1; out.z = r2; out.w = r3;
  *(float4*)(recon + (size_t)b * IN_DIM + t * 4) = out;
}

// ---------------------------------------------------------------------------
extern "C" void kernel_launch(void* const* d_in, const int* in_sizes, int n_in,
                              void* d_out, int out_size, void* d_ws, size_t ws_size,
                              hipStream_t stream) {
  const float* x    = (const float*)d_in[0];   // [8192,1024]
  const float* dict = (const float*)d_in[1];   // [1024,4096]
  const float* Ws   = (const float*)d_in[2];   // [5,1024,4096]
  const float* Ss   = (const float*)d_in[3];   // [5,4096,4096]

  float* recon = (float*)d_out;                               // [8192,1024]
  float* zOut  = (float*)d_out + (size_t)BATCH * IN_DIM;      // [8192,4096]

  char* ws = (char*)d_ws;
  float* P      = (float*)ws;                                             // 128 MB
  float* Dt     = (float*)(ws + (size_t)BATCH * CODE * 4);                //  16 MB
  int*   idxs   = (int*)  ((char*)Dt + (size_t)CODE * IN_DIM * 4);        //   2 MB
  float* vals   = (float*)((char*)idxs + (size_t)BATCH * KSP * 4);        //   2 MB
  int*   counts = (int*)  ((char*)vals + (size_t)BATCH * KSP * 4);        //  32 KB

  zero_counts_kernel<<<BATCH / 256, 256, 0, stream>>>(counts, BATCH);
  transpose_dict_kernel<<<dim3(CODE / 32, IN_DIM / 32), 256, 0, stream>>>(dict, Dt);

  for (int layer = 0; layer < NLAYERS; ++layer) {
    const float* W = Ws + (size_t)layer * IN_DIM * CODE;
    const float* S = Ss + (size_t)layer * CODE * CODE;
    // block tile 128x128 (8 waves of 64x32 macro-tiles)
    gemm_xw_wmma<<<dim3(CODE / 128, BATCH / 128), 256, 0, stream>>>(x, W, P);
    lista_step_kernel<<<BATCH, 256, 0, stream>>>(P, S, idxs, vals, counts, zOut,
                                                 layer == NLAYERS - 1 ? 1 : 0);
  }
  recon_kernel<<<BATCH, 256, 0, stream>>>(Dt, idxs, vals, counts, recon);
}